// VCStudentModel_72919954751585
// MI455X (gfx1250) — compile-verified
//
#include <hip/hip_runtime.h>
#include <hip/hip_bf16.h>

#define NUM_REGIONS 64
#define KNBR 5
#define BPTS 16384
#define ROWTILES (BPTS / 16)       // 1024 (one block per row tile)
#define COLTILES (BPTS / 16)       // 1024
#define WAVES_PER_BLOCK 4
#define TILES_PER_WAVE (COLTILES / WAVES_PER_BLOCK)  // 256

// ws layout (floats):
// [0..63]    region counts
// [64..255]  region sums   (r*3+d)
// [256..447] region sum-sq (r*3+d)
// [448]      spatial squared-diff accumulator
#define WS_COUNTS 0
#define WS_SUMS   64
#define WS_SQS    256
#define WS_SPAT   448
#define WS_TOTAL  449

typedef __attribute__((ext_vector_type(2))) float v2f;
typedef __attribute__((ext_vector_type(8))) float v8f;

__global__ void zero_ws_kernel(float* ws) {
    int i = blockIdx.x * blockDim.x + threadIdx.x;
    if (i < WS_TOTAL) ws[i] = 0.0f;
}

// ---- Region loss partial sums: LDS-staged segment reduction ----
__global__ void region_kernel(const float* __restrict__ cp,
                              const int* __restrict__ rid,
                              float* __restrict__ ws) {
    __shared__ float acc[NUM_REGIONS * 7]; // count, sum[3], sq[3] per region
    for (int i = threadIdx.x; i < NUM_REGIONS * 7; i += blockDim.x) acc[i] = 0.0f;
    __syncthreads();

    int i = blockIdx.x * blockDim.x + threadIdx.x;
    if (i < BPTS) {
        int r = rid[i];
        float x = cp[i * 3 + 0], y = cp[i * 3 + 1], z = cp[i * 3 + 2];
        atomicAdd(&acc[r * 7 + 0], 1.0f);
        atomicAdd(&acc[r * 7 + 1], x);
        atomicAdd(&acc[r * 7 + 2], y);
        atomicAdd(&acc[r * 7 + 3], z);
        atomicAdd(&acc[r * 7 + 4], x * x);
        atomicAdd(&acc[r * 7 + 5], y * y);
        atomicAdd(&acc[r * 7 + 6], z * z);
    }
    __syncthreads();

    for (int s = threadIdx.x; s < NUM_REGIONS * 7; s += blockDim.x) {
        float v = acc[s];
        if (v != 0.0f) {
            int r = s / 7, c = s % 7;
            float* dst = (c == 0) ? &ws[WS_COUNTS + r]
                       : (c <= 3) ? &ws[WS_SUMS + r * 3 + (c - 1)]
                                  : &ws[WS_SQS + r * 3 + (c - 4)];
            atomicAdd(dst, v);
        }
    }
}

// Branchless sorted insert (ascending) of (d, j) into a 5-entry list.
#define TOP5_INSERT(D0, D1, D2, D3, D4, I0, I1, I2, I3, I4, d, j)              \
    do {                                                                       \
        bool c0 = (d) < (D0), c1 = (d) < (D1), c2 = (d) < (D2),                \
             c3 = (d) < (D3), c4 = (d) < (D4);                                 \
        (D4) = c4 ? (c3 ? (D3) : (d)) : (D4);                                  \
        (I4) = c4 ? (c3 ? (I3) : (j)) : (I4);                                  \
        (D3) = c3 ? (c2 ? (D2) : (d)) : (D3);                                  \
        (I3) = c3 ? (c2 ? (I2) : (j)) : (I3);                                  \
        (D2) = c2 ? (c1 ? (D1) : (d)) : (D2);                                  \
        (I2) = c2 ? (c1 ? (I1) : (j)) : (I2);                                  \
        (D1) = c1 ? (c0 ? (D0) : (d)) : (D1);                                  \
        (I1) = c1 ? (c0 ? (I0) : (j)) : (I1);                                  \
        (D0) = c0 ? (d) : (D0);                                                \
        (I0) = c0 ? (j) : (I0);                                                \
    } while (0)

#define FBIG 3.402823466e+38f

// ---- KNN via V_WMMA_F32_16X16X4_F32 Gram tiles + register top-5 ----
// Block = 4 waves, all covering the SAME 16 query rows, interleaved over
// column tiles (wave w takes tiles t = w, w+4, w+8, ...). Per-row merge of
// the 4*16 lane-partial lists happens in LDS at the end.
__global__ void __launch_bounds__(32 * WAVES_PER_BLOCK)
knn_spatial_kernel(const float* __restrict__ cp,     // c_points [B,3]
                   const float* __restrict__ cs,     // coords_std [B,3]
                   float* __restrict__ ws) {
    const int lane = threadIdx.x & 31;
    const int wave = threadIdx.x >> 5;
    const int rowBase = blockIdx.x * 16;
    const int hi = lane >> 4;       // 0: K=0,1 / M=0..7 ; 1: K=2,3 / M=8..15
    const int n  = lane & 15;       // column within tile / row within A frag

    // A fragment: query coords. Lane holds row (rowBase+n): (x,y) low half, (z,0) high half.
    const int qi = rowBase + n;
    float qx = cs[qi * 3 + 0], qy = cs[qi * 3 + 1], qz = cs[qi * 3 + 2];
    v2f afrag;
    afrag.x = hi ? qz : qx;
    afrag.y = hi ? 0.0f : qy;

    // Squared norms for the 8 C-slot rows this lane produces: rowBase + v + 8*hi.
    float q2[8];
#pragma unroll
    for (int v = 0; v < 8; ++v) {
        int r = rowBase + v + 8 * hi;
        float x = cs[r * 3 + 0], y = cs[r * 3 + 1], z = cs[r * 3 + 2];
        q2[v] = x * x + y * y + z * z;
    }

    // Running per-slot top-5 (sorted ascending), register-resident.
    float bD[8][5];
    int   bI[8][5];
#pragma unroll
    for (int v = 0; v < 8; ++v)
#pragma unroll
        for (int k = 0; k < 5; ++k) { bD[v][k] = FBIG; bI[v][k] = 0; }

    for (int s = 0; s < TILES_PER_WAVE; ++s) {
        const int t = s * WAVES_PER_BLOCK + wave;   // interleaved column tiles
        const int pj = t * 16 + n;
        float px = cs[pj * 3 + 0], py = cs[pj * 3 + 1], pz = cs[pj * 3 + 2];
        v2f bfrag;
        bfrag.x = hi ? pz : px;
        bfrag.y = hi ? 0.0f : py;
        float p2 = px * px + py * py + pz * pz;

        v8f c = {};
        c = __builtin_amdgcn_wmma_f32_16x16x4_f32(
            /*neg_a=*/false, afrag, /*neg_b=*/false, bfrag,
            /*c_mod=*/(short)0, c, /*reuse_a=*/false, /*reuse_b=*/false);

#pragma unroll
        for (int v = 0; v < 8; ++v) {
            int rowIdx = rowBase + v + 8 * hi;
            float d2 = q2[v] + p2 - 2.0f * c[v];
            d2 = (pj == rowIdx) ? FBIG : d2;   // exclude self
            // Wave-uniform early-out: skip the insert network when no lane improves.
            if (__any(d2 < bD[v][4])) {
                TOP5_INSERT(bD[v][0], bD[v][1], bD[v][2], bD[v][3], bD[v][4],
                            bI[v][0], bI[v][1], bI[v][2], bI[v][3], bI[v][4],
                            d2, pj);
            }
        }
    }

    // Merge the 64 (wave, lane-column) partials per row through LDS.
    __shared__ float sD[16][WAVES_PER_BLOCK * 16][5];
    __shared__ int   sI[16][WAVES_PER_BLOCK * 16][5];
    __shared__ float spart[16];
#pragma unroll
    for (int v = 0; v < 8; ++v) {
        int row = v + 8 * hi;
        int col = wave * 16 + n;
#pragma unroll
        for (int k = 0; k < 5; ++k) {
            sD[row][col][k] = bD[v][k];
            sI[row][col][k] = bI[v][k];
        }
    }
    __syncthreads();

    if (threadIdx.x < 16) {
        const int m = threadIdx.x;       // row within the tile
        float fD0 = FBIG, fD1 = FBIG, fD2 = FBIG, fD3 = FBIG, fD4 = FBIG;
        int   fI0 = 0, fI1 = 0, fI2 = 0, fI3 = 0, fI4 = 0;
        for (int col = 0; col < WAVES_PER_BLOCK * 16; ++col) {
#pragma unroll
            for (int k = 0; k < 5; ++k) {
                float d = sD[m][col][k];
                int   j = sI[m][col][k];
                TOP5_INSERT(fD0, fD1, fD2, fD3, fD4, fI0, fI1, fI2, fI3, fI4, d, j);
            }
        }
        // Spatial contribution: sum_k ||c_i - c_{nb_k}||^2
        const int i = rowBase + m;
        float cx = cp[i * 3 + 0], cy = cp[i * 3 + 1], cz = cp[i * 3 + 2];
        float ssum = 0.0f;
        int idxs[5] = {fI0, fI1, fI2, fI3, fI4};
#pragma unroll
        for (int k = 0; k < 5; ++k) {
            int j = idxs[k];
            float dx = cx - cp[j * 3 + 0];
            float dy = cy - cp[j * 3 + 1];
            float dz = cz - cp[j * 3 + 2];
            ssum += dx * dx + dy * dy + dz * dz;
        }
        spart[m] = ssum;
    }
    __syncthreads();
    if (threadIdx.x == 0) {
        float blockSum = 0.0f;
#pragma unroll
        for (int m = 0; m < 16; ++m) blockSum += spart[m];
        atomicAdd(&ws[WS_SPAT], blockSum);   // one atomic per block
    }
}

// ---- Finalize: region loss reduction + spatial scaling ----
__global__ void finalize_kernel(const float* __restrict__ ws, float* __restrict__ out) {
    __shared__ float red[NUM_REGIONS];
    int r = threadIdx.x;
    float nr = ws[WS_COUNTS + r];
    float safe_n = fmaxf(nr, 1.0f);
    float ssc = 0.0f;
#pragma unroll
    for (int d = 0; d < 3; ++d) {
        float s = ws[WS_SUMS + r * 3 + d];
        float q = ws[WS_SQS + r * 3 + d];
        float m = s / safe_n;
        ssc += q - nr * m * m;
    }
    float npairs = nr * (nr - 1.0f) * 0.5f;
    float denom = fmaxf(npairs, 1.0f) * 3.0f;
    float pr = (nr > 1.0f) ? (2.0f * ssc / denom) : 0.0f;
    red[r] = pr;
    __syncthreads();
    for (int s = NUM_REGIONS / 2; s > 0; s >>= 1) {
        if (r < s) red[r] += red[r + s];
        __syncthreads();
    }
    if (r == 0) {
        float region_loss = 0.02f * red[0];
        float spatial = ws[WS_SPAT] / ((float)BPTS * (float)KNBR);   // mean over (B,k)
        float spatial_loss = 0.0005f * (spatial / 3.0f);             // / xdim
        out[0] = region_loss + spatial_loss;
    }
}

extern "C" void kernel_launch(void* const* d_in, const int* in_sizes, int n_in,
                              void* d_out, int out_size, void* d_ws, size_t ws_size,
                              hipStream_t stream) {
    const float* cp  = (const float*)d_in[0];  // c_points   [B,3]
    const float* cs  = (const float*)d_in[1];  // coords_std [B,3]
    const int*   rid = (const int*)d_in[2];    // region_ids [B]
    float* out = (float*)d_out;
    float* ws  = (float*)d_ws;

    zero_ws_kernel<<<2, 256, 0, stream>>>(ws);
    region_kernel<<<BPTS / 256, 256, 0, stream>>>(cp, rid, ws);
    knn_spatial_kernel<<<ROWTILES, 32 * WAVES_PER_BLOCK, 0, stream>>>(cp, cs, ws);
    finalize_kernel<<<1, NUM_REGIONS, 0, stream>>>(ws, out);
}